// Embeddings_84868553769453
// MI455X (gfx1250) — compile-verified
//
#include <hip/hip_runtime.h>

// Problem constants (from reference)
#define B_      64
#define S_      64
#define W_      16
#define CVOCAB  128
#define CDIM    64
#define OUTCH   256
#define WDIM    300
#define ODIM    556          // WDIM + OUTCH
#define NROWS   4096         // B_*S_
#define Q_PER_ROW 75         // 300 floats = 75 float4

// char kernel config
#define CBLOCKS        256
#define PAIRS_PER_BLK  (NROWS / CBLOCKS)   // 16
#define PQ             4                   // pairs in flight per iteration

// Native 4 x f32 vector (16B aligned) — accepted by nontemporal builtins and
// lowered to b128 loads/stores.
typedef float f4 __attribute__((ext_vector_type(4)));

// Low 32 bits of a generic pointer to a __shared__ object are the LDS byte
// offset (flat LDS aperture: ADDR[31:0] == LDS offset, see CDNA5 ISA 10.2).
__device__ __forceinline__ unsigned lds_off(const void* p) {
    return (unsigned)(unsigned long long)p;
}

// CDNA5 async global->LDS copy, 16B per lane. Tracked with ASYNCcnt.
__device__ __forceinline__ void async_copy_b128(unsigned lds_addr,
                                                unsigned long long gaddr) {
    asm volatile("global_load_async_to_lds_b128 %0, %1, off"
                 :: "v"(lds_addr), "v"(gaddr)
                 : "memory");
}

__device__ __forceinline__ void wait_asynccnt0() {
    asm volatile("s_wait_asynccnt 0x0" ::: "memory");
}

// ---------------------------------------------------------------------------
// Kernel A (defined first so its disasm leads the object): char embedding
// gather + depthwise conv(K=3, 4 filters/channel) + bias + max over 14
// positions, written transposed:
//   chars row n = s*64+b  ->  out[(b*64+s)*556 + 300 + oc]
// The 32KB char table, 3KB weights and 1KB bias are staged into LDS with
// global_load_async_to_lds_b128 and all per-pair gathers are LDS-resident.
// ---------------------------------------------------------------------------
__global__ void __launch_bounds__(256)
char_cnn_kernel(const int* __restrict__ chars,
                const float* __restrict__ cemb,
                const float* __restrict__ cw,
                const float* __restrict__ cb,
                float* __restrict__ out) {
    __shared__ __align__(16) float table[CVOCAB * CDIM];   // 8192 f32 = 32KB
    __shared__ __align__(16) float wts[OUTCH * 3];         // 768 f32  =  3KB
    __shared__ __align__(16) float bias[OUTCH];            // 256 f32  =  1KB
    __shared__ __align__(16) float xs[PQ][W_][CDIM];       // 4096 f32 = 16KB
    __shared__ int cidx[PQ][W_];

    const int tid = threadIdx.x;

    // ---- async stage: char table (8 x b128 per lane), weights, bias ----
    {
        const unsigned tl = lds_off(&table[0]);
        const unsigned long long tg = (unsigned long long)(const void*)cemb;
        #pragma unroll
        for (int k = 0; k < 8; ++k) {
            unsigned e = (unsigned)(tid + k * 256);
            async_copy_b128(tl + e * 16u, tg + (unsigned long long)e * 16ull);
        }
        if (tid < 192) {   // 768 floats = 192 float4
            async_copy_b128(lds_off(&wts[0]) + (unsigned)tid * 16u,
                            (unsigned long long)(const void*)cw +
                                (unsigned long long)tid * 16ull);
        } else {           // 256 floats = 64 float4
            unsigned e = (unsigned)(tid - 192);
            async_copy_b128(lds_off(&bias[0]) + e * 16u,
                            (unsigned long long)(const void*)cb +
                                (unsigned long long)e * 16ull);
        }
        wait_asynccnt0();   // this wave's ASYNCcnt == 0 -> its LDS writes done
    }
    __syncthreads();        // make all waves' staged data visible

    const int pt    = tid >> 6;    // pair-in-flight 0..3
    const int lane6 = tid & 63;    // worker id within pair == input channel c
    const int base  = blockIdx.x * PAIRS_PER_BLK;

    for (int p = 0; p < PAIRS_PER_BLK; p += PQ) {
        // 64 char indices (4 pairs x 16) by the first 64 threads
        if (tid < PQ * W_) {
            int pp = tid >> 4, ww = tid & 15;
            cidx[pp][ww] = chars[(base + p + pp) * W_ + ww];
        }
        __syncthreads();

        // gather char embeddings into xs: 1024 f32/pair, 64 lanes/pair,
        // 4 x b128 LDS->LDS moves per lane, bank-spread across c.
        {
            const int c4 = lane6 & 15;   // float4 column 0..15
            const int wg = lane6 >> 4;   // word-group 0..3
            const f4* t4 = (const f4*)table;
            f4*       x4 = (f4*)&xs[pt][0][0];
            #pragma unroll
            for (int j = 0; j < 4; ++j) {
                int ww = wg * 4 + j;
                x4[ww * (CDIM / 4) + c4] = t4[cidx[pt][ww] * (CDIM / 4) + c4];
            }
        }
        __syncthreads();

        // depthwise conv + bias + max.  lane -> channel c; computes the 4
        // output channels of its group from 16 register-resident x values.
        {
            const int c = lane6;
            float x[W_];
            #pragma unroll
            for (int ww = 0; ww < W_; ++ww) x[ww] = xs[pt][ww][c];

            const int n    = base + p + pt;            // chars row s*64+b
            const int orow = (n & 63) * 64 + (n >> 6); // out row  b*64+s

            f4 rv;
            #pragma unroll
            for (int f = 0; f < 4; ++f) {
                const int oc = c * 4 + f;
                const float w0 = wts[oc * 3 + 0];
                const float w1 = wts[oc * 3 + 1];
                const float w2 = wts[oc * 3 + 2];
                float m = -3.402823466e+38f;
                #pragma unroll
                for (int t = 0; t <= W_ - 3; ++t) {
                    float v = fmaf(w2, x[t + 2],
                              fmaf(w1, x[t + 1], w0 * x[t]));
                    m = fmaxf(m, v);
                }
                rv[f] = m + bias[oc];
            }
            // streaming, write-once output -> non-temporal b128 store
            __builtin_nontemporal_store(
                rv, (f4*)(out + (long long)orow * ODIM + WDIM + c * 4));
        }
        __syncthreads();   // protect cidx/xs before next iteration
    }
}

// ---------------------------------------------------------------------------
// Kernel B: word embedding gather  out[row*556 + 0..299] = wemb[words[row]]
// One float4 per thread; 4096 rows * 75 float4 = 307200 threads exactly.
// Gathered rows stay RT-cached in L2 (word indices can repeat); the output
// store is non-temporal (write-once stream).
// ---------------------------------------------------------------------------
__global__ void __launch_bounds__(256)
word_gather_kernel(const int* __restrict__ words,
                   const float* __restrict__ wemb,
                   float* __restrict__ out) {
    int i   = blockIdx.x * blockDim.x + threadIdx.x;   // 0..307199
    int row = i / Q_PER_ROW;
    int q   = i - row * Q_PER_ROW;
    int idx = words[row];
    const f4* src = (const f4*)(wemb + (long long)idx * WDIM);
    f4*       dst = (f4*)(out + (long long)row * ODIM);
    __builtin_nontemporal_store(src[q], dst + q);
}

// ---------------------------------------------------------------------------
extern "C" void kernel_launch(void* const* d_in, const int* in_sizes, int n_in,
                              void* d_out, int out_size, void* d_ws, size_t ws_size,
                              hipStream_t stream) {
    const int*   words = (const int*)d_in[0];   // [64,64] int32
    const int*   chars = (const int*)d_in[1];   // [64,64,16] int32
    const float* wemb  = (const float*)d_in[2]; // [128000,300] f32
    const float* cemb  = (const float*)d_in[3]; // [128,64] f32
    const float* cw    = (const float*)d_in[4]; // [256,1,3] f32
    const float* cb    = (const float*)d_in[5]; // [256] f32
    float*       out   = (float*)d_out;         // [64,64,556] f32

    char_cnn_kernel<<<CBLOCKS, 256, 0, stream>>>(chars, cemb, cw, cb, out);

    // 4096 rows * 75 float4 = 307200 threads exactly
    word_gather_kernel<<<(NROWS * Q_PER_ROW) / 256, 256, 0, stream>>>(
        words, wemb, out);
}